// CausalProjectionLayer_86672440033699
// MI455X (gfx1250) — compile-verified
//
#include <hip/hip_runtime.h>
#include <hip/hip_bf16.h>
#include <stdint.h>

// ---------------------------------------------------------------------------
// CausalProjectionLayer fused kernel for gfx1250 (MI455X).
//
// Reference broadcasting: A[j-1] indexes the *L* axis (L == D == 384),
// delta_t indexes the *B* axis. 8 sequential shift-stencil steps:
//   x[b,l,d] = x[b,l,d] + A[j-1][l]*dx + A'[j-1][l]*(dx/dt[b]),
//   dx = x[b,l,d] - x[b,l-j,d]  (0 when l < j)
//
// Memory-bound: 216 MB in + 216 MB out fused over all 8 steps (~19 us at
// 23.3 TB/s). Each thread keeps a 24-long l-strip of one (b,d) column in
// registers; only <=8 boundary values per step cross threads via a small LDS
// slot array. Coefficients are staged into LDS via async global->LDS b128 and
// folded once per block into c[j][l] = A[j][l] + A'[j][l]*invdt (invdt==1.0
// exactly), so the per-element-step body is just sub + FMA (2 VALU ops),
// keeping VALU (~11 us) under the HBM roofline (~19 us).
// ---------------------------------------------------------------------------

#define AS1 __attribute__((address_space(1)))
#define AS3 __attribute__((address_space(3)))

typedef int v4i __attribute__((ext_vector_type(4)));

#if defined(__has_builtin)
#  if __has_builtin(__builtin_amdgcn_global_load_async_to_lds_b128)
#    define HAVE_ASYNC_B128 1
#  endif
#  if __has_builtin(__builtin_amdgcn_s_wait_asynccnt)
#    define HAVE_WAIT_ASYNC 1
#  endif
#endif

__device__ __forceinline__ void async_copy_b128(float* lds_dst, const float* gsrc) {
#if defined(HAVE_ASYNC_B128)
  // async global -> LDS, tracked with ASYNCcnt (no VGPR round trip)
  __builtin_amdgcn_global_load_async_to_lds_b128((AS1 v4i*)gsrc, (AS3 v4i*)lds_dst, 0, 0);
#else
  *reinterpret_cast<float4*>(lds_dst) = *reinterpret_cast<const float4*>(gsrc);
#endif
}

__device__ __forceinline__ void wait_async_lds() {
#if defined(HAVE_ASYNC_B128)
#  if defined(HAVE_WAIT_ASYNC)
  __builtin_amdgcn_s_wait_asynccnt(0);
#  else
  asm volatile("s_wait_asynccnt 0" ::: "memory");
#  endif
#endif
}

namespace cproj {

constexpr int B  = 384;
constexpr int L  = 384;
constexpr int D  = 384;
constexpr int P  = 8;

constexpr int TD = 32;        // d-columns per block (one wave spans them)
constexpr int NS = 16;        // l-strips per block
constexpr int LT = L / NS;    // 24 l-values held in registers per thread
constexpr int NT = TD * NS;   // 512 threads per block

static_assert(L % NS == 0, "strip split");
static_assert(LT >= P, "strip must cover max shift");
static_assert((P * L) % 4 == 0, "vec4 coeff load");

__global__ __launch_bounds__(NT)
void causal_proj_kernel(const float* __restrict__ x,
                        const float* __restrict__ A,
                        const float* __restrict__ Ap,
                        const float* __restrict__ dtv,
                        float* __restrict__ out) {
  __shared__ float sA [P * L];           // 12 KB  A[j][l]   (async-staged)
  __shared__ float sAp[P * L];           // 12 KB  A'[j][l]  (async-staged)
  __shared__ float sC [P * L];           // 12 KB  folded c[j][l]
  __shared__ float slots[NS * P * TD];   // 16 KB  boundary exchange

  const int tid   = threadIdx.x;
  const int d     = tid & (TD - 1);
  const int strip = tid >> 5;            // wave-uniform (wave32, TD==32)
  const int b     = blockIdx.y;
  const int d0    = blockIdx.x * TD;
  const int l0    = strip * LT;

  // ---- stage coefficient tables into LDS with async global->LDS b128 ----
  {
    const int nv4 = (P * L) / 4;         // 768 float4 per table
    for (int e = tid; e < nv4; e += NT) async_copy_b128(&sA [e * 4], A  + e * 4);
    for (int e = tid; e < nv4; e += NT) async_copy_b128(&sAp[e * 4], Ap + e * 4);
  }

  // ---- stream this thread's column strip straight into registers ----
  // consecutive lanes hit consecutive d -> 128B coalesced per l; nontemporal
  // (x is 216 MB read-once: don't thrash the 192 MB L2)
  const float* __restrict__ xg = x + ((size_t)b * L + l0) * D + d0 + d;
  float v[LT];
#pragma unroll
  for (int i = 0; i < LT; ++i)
    v[i] = __builtin_nontemporal_load(xg + (size_t)i * D);

  const float invdt = 1.0f / dtv[b];     // delta_t broadcasts along B; == 1.0

  wait_async_lds();
  __syncthreads();

  // ---- fold c[j][l] = A[j][l] + A'[j][l]*invdt once per block ----
  // invdt == 1.0 exactly, so A'[j][l]*invdt == A'[j][l] bitwise; folding
  // differs from the reference only by a single fp32 rounding per step.
  for (int i = tid; i < P * L; i += NT)
    sC[i] = __builtin_fmaf(sAp[i], invdt, sA[i]);
  __syncthreads();

  // ---- 8 fused recurrence steps, values resident in registers ----
#pragma unroll
  for (int j = 1; j <= P; ++j) {
    // publish this strip's top-j (pre-update) values for the strip above
#pragma unroll
    for (int t = 0; t < j; ++t)
      slots[(strip * P + t) * TD + d] = v[LT - j + t];
    __syncthreads();

    float nb[P];
    if (strip > 0) {                     // wave-uniform branch
#pragma unroll
      for (int t = 0; t < j; ++t)
        nb[t] = slots[((strip - 1) * P + t) * TD + d];
    }

    const float* __restrict__ cj = &sC[(j - 1) * L + l0];  // broadcast reads

    // in-place, descending i: v[i-j] is still the previous step's value
#pragma unroll
    for (int i = LT - 1; i >= j; --i) {
      float dx = v[i] - v[i - j];
      v[i] = __builtin_fmaf(cj[i], dx, v[i]);
    }
    if (strip > 0) {
#pragma unroll
      for (int i = P - 1; i >= 0; --i) {
        if (i < j) {
          float dx = v[i] - nb[i];
          v[i] = __builtin_fmaf(cj[i], dx, v[i]);
        }
      }
    }
    // else: l = i < j  ==> masked, dx = 0, x unchanged (exact)
    __syncthreads();
  }

  // ---- stream results back (write-once: nontemporal) ----
  float* __restrict__ og = out + ((size_t)b * L + l0) * D + d0 + d;
#pragma unroll
  for (int i = 0; i < LT; ++i)
    __builtin_nontemporal_store(v[i], og + (size_t)i * D);
}

}  // namespace cproj

extern "C" void kernel_launch(void* const* d_in, const int* in_sizes, int n_in,
                              void* d_out, int out_size, void* d_ws, size_t ws_size,
                              hipStream_t stream) {
  (void)in_sizes; (void)n_in; (void)out_size; (void)d_ws; (void)ws_size;
  const float* x   = (const float*)d_in[0];   // (B, L, D) f32
  const float* A   = (const float*)d_in[1];   // (P, D)    f32
  const float* Ap  = (const float*)d_in[2];   // (P, D)    f32
  const float* dtv = (const float*)d_in[3];   // (L, 1)    f32
  float* out = (float*)d_out;                 // (B, L, D) f32

  dim3 grid(cproj::D / cproj::TD, cproj::B, 1);   // (12, 384)
  dim3 block(cproj::NT, 1, 1);                    // 512 threads = 16 waves
  cproj::causal_proj_kernel<<<grid, block, 0, stream>>>(x, A, Ap, dtv, out);
}